// ExactModel_9586367004881
// MI455X (gfx1250) — compile-verified
//
#include <hip/hip_runtime.h>
#include <math.h>

// ---------------------------------------------------------------------------
// out[i] = sum_j tanh(K*(S[i]-S[j]) - EPS), S from segment logsumexp + dot.
//
// Phase-2 dominates: N^2 = 67M tanh + 67M accumulations, ~0 HBM traffic.
//  - tanh  -> native v_tanh_f32 (TRANS pipe), 2 per asm block so the second
//    covers the first's TRANS hazard (1 trailing v_nop per pair)
//  - row-reduction -> V_WMMA_F32_16X16X4_F32 with all-ones B
//    (D = Tanh_tile(16x4) * ones(4x16) + D), tanh values produced directly
//    in the f32 A-matrix VGPR layout (2 per lane)
//  - 4x ds_load_b64 batched per iteration so one dscnt wait covers 16 j's
// ---------------------------------------------------------------------------

typedef float v2f __attribute__((ext_vector_type(2)));
typedef float v8f __attribute__((ext_vector_type(8)));

#define K_SIGN   1000.0f
#define EPSILON  5.0f
#define JCHUNK   512

// two tanh in one block: 2nd tanh is the independent op covering the 1st's
// TRANS RAW hazard; v_nop covers the 2nd. Non-volatile so the scheduler can
// hoist LDS loads across it.
__device__ __forceinline__ v2f tanh2_hw(float x0, float x1) {
  float r0, r1;
  asm("v_tanh_f32 %0, %2\n\t"
      "v_tanh_f32 %1, %3\n\t"
      "v_nop"
      : "=&v"(r0), "=v"(r1)
      : "v"(x0), "v"(x1));
  v2f a; a.x = r0; a.y = r1;
  return a;
}

// ---- phase 0: init workspace + output -------------------------------------
__global__ void k_init(unsigned* mx, float* se, float* dotacc, float* out, int n) {
  int i = blockIdx.x * blockDim.x + threadIdx.x;
  if (i < n) { mx[i] = 0u; se[i] = 0.0f; out[i] = 0.0f; }
  if (i == 0) *dotacc = 0.0f;
}

// ---- phase 1a: segment max of contrib = log(p[src]) + p[dst]*log(n) -------
// contribs are strictly positive -> uint compare == float compare.
__global__ void k_edge_max(const int* src, const int* dst, const float* p,
                           unsigned* mx, int E, int n, float log_n) {
  int g = blockIdx.x * blockDim.x + threadIdx.x;
  if (g >= E + n) return;
  int s, d;
  if (g < E) { s = src[g]; d = dst[g]; } else { s = d = g - E; }
  float c = logf(p[s]) + p[d] * log_n;
  atomicMax(&mx[d], __float_as_uint(c));
}

// ---- phase 1b: segment sum of exp(contrib - mx[dst]) ----------------------
__global__ void k_edge_sum(const int* src, const int* dst, const float* p,
                           const unsigned* mx, float* se, int E, int n, float log_n) {
  int g = blockIdx.x * blockDim.x + threadIdx.x;
  if (g >= E + n) return;
  int s, d;
  if (g < E) { s = src[g]; d = dst[g]; } else { s = d = g - E; }
  float c = logf(p[s]) + p[d] * log_n;
  atomicAdd(&se[d], expf(c - __uint_as_float(mx[d])));
}

// ---- phase 1c: scalar dot(x, p) via wave32 shuffle reduction --------------
__global__ void k_dot(const float* x, const float* p, float* dotacc, int n) {
  int i = blockIdx.x * blockDim.x + threadIdx.x;
  float v = (i < n) ? x[i] * p[i] : 0.0f;
  #pragma unroll
  for (int off = 16; off >= 1; off >>= 1) v += __shfl_xor(v, off, 32);
  if ((threadIdx.x & 31) == 0) atomicAdd(dotacc, v);
}

// ---- phase 1d: S[i] = mx[i] + log(se[i]) + dot ----------------------------
__global__ void k_S(const unsigned* mx, const float* se, const float* dotacc,
                    float* S, int n) {
  int i = blockIdx.x * blockDim.x + threadIdx.x;
  if (i < n) S[i] = __uint_as_float(mx[i]) + logf(se[i]) + *dotacc;
}

// ---- phase 2: pairwise tanh + WMMA row-sum --------------------------------
// One wave per block. Block handles 16 rows x one JCHUNK of columns.
// A-matrix layout (f32 16x4): lane L -> row M = L&15, K = 2*(L>>4) + {0,1}.
// C/D layout: VGPR r -> row r (lanes 0-15) / row r+8 (lanes 16-31).
__global__ void __launch_bounds__(32)
k_pairwise(const float* __restrict__ S, float* __restrict__ out, int n) {
  __shared__ float sj[JCHUNK];

  const int lane  = threadIdx.x;
  const int ib    = blockIdx.x * 16;
  const int jbase = blockIdx.y * JCHUNK;
  const bool full = (jbase + JCHUNK <= n);

  // stage this block's column chunk of S into LDS
  if (full) {
    const float4* Sv  = (const float4*)(S + jbase);   // jbase is 512-aligned
    float4*       sjv = (float4*)sj;
    #pragma unroll
    for (int t = lane; t < JCHUNK / 4; t += 32) sjv[t] = Sv[t];
  } else {
    for (int t = lane; t < JCHUNK; t += 32) {
      int j = jbase + t;
      sj[t] = (j < n) ? S[j] : 0.0f;
    }
  }
  __syncthreads();

  const int   row = lane & 15;
  const int   kg  = (lane >> 4) << 1;            // K offset: 0 or 2
  const int   i   = ib + row;
  const float si  = (i < n) ? S[i] : 0.0f;
  // K*(si - sj) - EPS == fma(-K, sj, pre) with pre = K*si - EPS
  const float pre = fmaf(K_SIGN, si, -EPSILON);

  v2f ones; ones.x = 1.0f; ones.y = 1.0f;        // B = all-ones 4x16
  v8f acc = {};

  const v2f* sj2 = (const v2f*)sj;               // 8B-aligned pair view

  if (full) {
    // ---- fast path: 16 j's per iteration, 4 batched ds_load_b64 ----
    for (int jo = 0; jo < JCHUNK; jo += 16) {
      const int t2 = (jo + kg) >> 1;             // float2 index
      v2f q0 = sj2[t2];
      v2f q1 = sj2[t2 + 2];
      v2f q2 = sj2[t2 + 4];
      v2f q3 = sj2[t2 + 6];

      v2f a0 = tanh2_hw(fmaf(-K_SIGN, q0.x, pre), fmaf(-K_SIGN, q0.y, pre));
      acc = __builtin_amdgcn_wmma_f32_16x16x4_f32(
          false, a0, false, ones, (short)0, acc, false, false);
      v2f a1 = tanh2_hw(fmaf(-K_SIGN, q1.x, pre), fmaf(-K_SIGN, q1.y, pre));
      acc = __builtin_amdgcn_wmma_f32_16x16x4_f32(
          false, a1, false, ones, (short)0, acc, false, false);
      v2f a2 = tanh2_hw(fmaf(-K_SIGN, q2.x, pre), fmaf(-K_SIGN, q2.y, pre));
      acc = __builtin_amdgcn_wmma_f32_16x16x4_f32(
          false, a2, false, ones, (short)0, acc, false, false);
      v2f a3 = tanh2_hw(fmaf(-K_SIGN, q3.x, pre), fmaf(-K_SIGN, q3.y, pre));
      acc = __builtin_amdgcn_wmma_f32_16x16x4_f32(
          false, a3, false, ones, (short)0, acc, false, false);
    }
  } else {
    // ---- tail path (unused at N=8192): guard out-of-range columns ----
    for (int jo = 0; jo < JCHUNK; jo += 4) {
      const int t = jo + kg;
      v2f q = sj2[t >> 1];
      v2f a = tanh2_hw(fmaf(-K_SIGN, q.x, pre), fmaf(-K_SIGN, q.y, pre));
      if (jbase + t     >= n) a.x = 0.0f;
      if (jbase + t + 1 >= n) a.y = 0.0f;
      acc = __builtin_amdgcn_wmma_f32_16x16x4_f32(
          false, a, false, ones, (short)0, acc, false, false);
    }
  }

  // column 0 lanes (lane 0 and lane 16) publish 8 row sums each
  if (row == 0) {
    const int mb = ib + ((lane >> 4) << 3);
    #pragma unroll
    for (int r = 0; r < 8; ++r) {
      if (mb + r < n) atomicAdd(&out[mb + r], acc[r]);
    }
  }
}

// ---------------------------------------------------------------------------
extern "C" void kernel_launch(void* const* d_in, const int* in_sizes, int n_in,
                              void* d_out, int out_size, void* d_ws, size_t ws_size,
                              hipStream_t stream) {
  const int*   edge = (const int*)d_in[0];   // (2,E) flat: [0..E)=src, [E..2E)=dst
  const float* p    = (const float*)d_in[1]; // (N,)
  const float* x    = (const float*)d_in[2]; // (N,1) flat

  const int E = in_sizes[0] / 2;
  const int n = in_sizes[1];
  float* out = (float*)d_out;

  // workspace layout: mx[n] (uint bits) | se[n] | S[n] | dotacc[1]
  unsigned* mx     = (unsigned*)d_ws;
  float*    se     = (float*)(mx + n);
  float*    S      = se + n;
  float*    dotacc = S + n;

  const float log_n = logf((float)n);
  const int tpb = 256;
  const int total = E + n;

  k_init    <<<(n + tpb - 1) / tpb, tpb, 0, stream>>>(mx, se, dotacc, out, n);
  k_edge_max<<<(total + tpb - 1) / tpb, tpb, 0, stream>>>(edge, edge + E, p, mx, E, n, log_n);
  k_edge_sum<<<(total + tpb - 1) / tpb, tpb, 0, stream>>>(edge, edge + E, p, mx, se, E, n, log_n);
  k_dot     <<<(n + tpb - 1) / tpb, tpb, 0, stream>>>(x, p, dotacc, n);
  k_S       <<<(n + tpb - 1) / tpb, tpb, 0, stream>>>(mx, se, dotacc, S, n);

  dim3 grid((n + 15) / 16, (n + JCHUNK - 1) / JCHUNK);
  k_pairwise<<<grid, 32, 0, stream>>>(S, out, n);
}